// ExphormerFullLayer_18786186952963
// MI455X (gfx1250) — compile-verified
//
#include <hip/hip_runtime.h>
#include <math.h>

typedef float v2f __attribute__((ext_vector_type(2)));
typedef float v8f __attribute__((ext_vector_type(8)));

#define D   128
#define NH  8
#define DH  16

#define KC      64            // K-chunk staged in LDS
#define KPAD    (KC + 2)      // padded stride (even -> 8B-aligned b64 reads)
#define WAVES   8             // waves (row tiles) per workgroup

// ---------------------------------------------------------------------------
// WMMA fp32 GEMM: C[M x ldb] = A[M x K] @ B[K x ldb] (+bias, +relu)
// 256 threads = 8 waves; each wave owns a 16-row x 128-col tile.
// B is staged transposed in LDS in 64-deep K chunks; each WMMA B-fragment is
// a single ds_load_b64.  V_WMMA_F32_16X16X4_F32 accumulates K in steps of 4.
// A-fragment (16x4 f32): lane L holds row m0+(L&15), k-pair k0+(L>>4)*2.
// C/D (16x16 f32): vgpr r holds row m0+r+(L>>4)*8, col n0+(L&15).
// M multiple of 16 (50000, 600000 ok); K in {128,256}; ldb in {128,256}.
// ---------------------------------------------------------------------------
__global__ __launch_bounds__(256) void gemm_wmma_f32(
    const float* __restrict__ A, const float* __restrict__ B,
    const float* __restrict__ bias, float* __restrict__ C,
    int M, int K, int ldb, int colBase, int relu)
{
  __shared__ float Bt[128 * KPAD];         // [col_local][k_in_chunk], padded

  const int tid   = threadIdx.x;
  const int lane  = tid & 31;
  const int wave  = tid >> 5;
  const int mtile = blockIdx.x * WAVES + wave;
  const int m0    = mtile * 16;
  const bool active = (m0 < M);            // wave-uniform

  const int lhalf = lane >> 4;             // 0 or 1
  const int lmod  = lane & 15;
  const int kh    = lhalf * 2;             // k-pair base within 4-wide step
  const size_t arow = (size_t)(m0 + lmod) * (size_t)K;

  // staging assignment: 256 threads cover 128 cols x 2 k-halves
  const int scol  = tid & 127;             // local column 0..127
  const int skoff = (tid >> 7) * (KC / 2); // 0 or 32

  v8f acc[8] = {};

  for (int kc = 0; kc < K; kc += KC) {
    // ---- stage B[kc..kc+63][colBase..colBase+127] transposed into LDS ----
#pragma unroll
    for (int i = 0; i < KC / 2; ++i) {
      const int k = skoff + i;
      Bt[scol * KPAD + k] = B[(size_t)(kc + k) * ldb + colBase + scol];
    }
    __syncthreads();

    if (active) {
      for (int k0 = 0; k0 < KC; k0 += 4) {
        v2f a = *(const v2f*)(A + arow + kc + k0 + kh);
#pragma unroll
        for (int t = 0; t < 8; ++t) {
          v2f b = *(const v2f*)(&Bt[(t * 16 + lmod) * KPAD + k0 + kh]);
          acc[t] = __builtin_amdgcn_wmma_f32_16x16x4_f32(
              false, a, false, b, (short)0, acc[t], false, false);
        }
      }
    }
    __syncthreads();
  }

  if (active) {
#pragma unroll
    for (int t = 0; t < 8; ++t) {
      const int col = colBase + t * 16 + lmod;
      const float bv = bias ? bias[col] : 0.0f;
#pragma unroll
      for (int r = 0; r < 8; ++r) {
        const int row = m0 + r + lhalf * 8;
        float v = acc[t][r] + bv;
        if (relu) v = fmaxf(v, 0.0f);
        C[(size_t)row * ldb + col] = v;
      }
    }
  }
}

// ---------------------------------------------------------------------------
// Edge attention: one thread per (edge, head).
// score = exp(clip(sum_d K[src]*Q[dst]*Ee / 4, -5, 5))
// scatter: wV[dst] += V[src]*score ; Z[dst,h] += score   (fp32 atomics, L2)
// ---------------------------------------------------------------------------
__global__ void edge_attn_scatter(
    const float* __restrict__ Q, const float* __restrict__ Kh,
    const float* __restrict__ V, const float* __restrict__ Ee,
    const int* __restrict__ ei, float* __restrict__ wV,
    float* __restrict__ Z, int E)
{
  long long idx = (long long)blockIdx.x * blockDim.x + threadIdx.x;
  if (idx >= (long long)E * NH) return;
  const int e = (int)(idx >> 3);
  const int h = (int)(idx & 7);
  const int s = ei[e];        // edge_index[0][e] = src
  const int d = ei[E + e];    // edge_index[1][e] = dst

  const float4* kp = (const float4*)(Kh + (size_t)s * D + h * DH);
  const float4* qp = (const float4*)(Q  + (size_t)d * D + h * DH);
  const float4* ep = (const float4*)(Ee + (size_t)e * D + h * DH);

  float sc = 0.0f;
#pragma unroll
  for (int i = 0; i < 4; ++i) {
    float4 k4 = kp[i], q4 = qp[i], e4 = ep[i];
    sc += k4.x * q4.x * e4.x + k4.y * q4.y * e4.y +
          k4.z * q4.z * e4.z + k4.w * q4.w * e4.w;
  }
  sc *= 0.25f;                                 // 1/sqrt(DH)
  sc = fminf(5.0f, fmaxf(-5.0f, sc));
  sc = expf(sc);

  const float4* vp = (const float4*)(V + (size_t)s * D + h * DH);
  float* out = wV + (size_t)d * D + h * DH;
#pragma unroll
  for (int i = 0; i < 4; ++i) {
    float4 v4 = vp[i];
    atomicAdd(out + i * 4 + 0, v4.x * sc);
    atomicAdd(out + i * 4 + 1, v4.y * sc);
    atomicAdd(out + i * 4 + 2, v4.z * sc);
    atomicAdd(out + i * 4 + 3, v4.w * sc);
  }
  atomicAdd(Z + (size_t)d * NH + h, sc);
}

// h1 = x + wV/(Z+1e-6); accumulate per-column sum/sumsq (128 thr = 128 cols)
__global__ void attn_residual_stats(
    const float* __restrict__ x, const float* __restrict__ wV,
    const float* __restrict__ Z, float* __restrict__ h1,
    float* __restrict__ sums, float* __restrict__ sumsq, int N)
{
  const int c = threadIdx.x;
  const int h = c >> 4;
  float s = 0.0f, s2 = 0.0f;
  for (int r = blockIdx.x; r < N; r += gridDim.x) {
    const size_t i = (size_t)r * D + c;
    float v = x[i] + wV[i] / (Z[(size_t)r * NH + h] + 1e-6f);
    h1[i] = v;
    s += v; s2 += v * v;
  }
  atomicAdd(&sums[c], s);
  atomicAdd(&sumsq[c], s2);
}

// out = a + b; accumulate per-column sum/sumsq
__global__ void add_residual_stats(
    const float* __restrict__ a, const float* __restrict__ b,
    float* __restrict__ out, float* __restrict__ sums,
    float* __restrict__ sumsq, int N)
{
  const int c = threadIdx.x;
  float s = 0.0f, s2 = 0.0f;
  for (int r = blockIdx.x; r < N; r += gridDim.x) {
    const size_t i = (size_t)r * D + c;
    float v = a[i] + b[i];
    out[i] = v;
    s += v; s2 += v * v;
  }
  atomicAdd(&sums[c], s);
  atomicAdd(&sumsq[c], s2);
}

// scale = gamma/sqrt(var+eps), shift = beta - mean*scale (biased var)
__global__ void bn_finalize(
    const float* __restrict__ sums, const float* __restrict__ sumsq,
    const float* __restrict__ gamma, const float* __restrict__ beta,
    float* __restrict__ scale, float* __restrict__ shift, int N)
{
  const int c = threadIdx.x;
  const float invN = 1.0f / (float)N;
  const float mean = sums[c] * invN;
  const float var  = sumsq[c] * invN - mean * mean;
  const float a    = gamma[c] / sqrtf(var + 1e-5f);
  scale[c] = a;
  shift[c] = beta[c] - mean * a;
}

__global__ void bn_apply(
    const float* __restrict__ h, const float* __restrict__ scale,
    const float* __restrict__ shift, float* __restrict__ out, size_t total)
{
  const size_t stride = (size_t)gridDim.x * blockDim.x;
  for (size_t i = (size_t)blockIdx.x * blockDim.x + threadIdx.x;
       i < total; i += stride) {
    const int c = (int)(i & (D - 1));
    out[i] = h[i] * scale[c] + shift[c];
  }
}

// ---------------------------------------------------------------------------
extern "C" void kernel_launch(void* const* d_in, const int* in_sizes, int n_in,
                              void* d_out, int out_size, void* d_ws, size_t ws_size,
                              hipStream_t stream)
{
  const float* x   = (const float*)d_in[0];
  const float* ea  = (const float*)d_in[1];
  const int*   ei  = (const int*)d_in[2];
  const float* Wq  = (const float*)d_in[3];
  const float* Wk  = (const float*)d_in[4];
  const float* We  = (const float*)d_in[5];
  const float* Wv  = (const float*)d_in[6];
  const float* g1  = (const float*)d_in[7];
  const float* be1 = (const float*)d_in[8];
  const float* W1  = (const float*)d_in[9];
  const float* b1  = (const float*)d_in[10];
  const float* W2  = (const float*)d_in[11];
  const float* b2  = (const float*)d_in[12];
  const float* g2  = (const float*)d_in[13];
  const float* be2 = (const float*)d_in[14];

  const int N = in_sizes[0] / D;
  const int E = in_sizes[1] / D;

  // workspace partition (all section sizes are multiples of 16 floats)
  float* ws  = (float*)d_ws;
  float* Qb  = ws;  ws += (size_t)N * D;
  float* Kb  = ws;  ws += (size_t)N * D;
  float* Vb  = ws;  ws += (size_t)N * D;
  float* wV  = ws;  ws += (size_t)N * D;
  float* Zb  = ws;  ws += (size_t)N * NH;
  float* h1  = ws;  ws += (size_t)N * D;     // reused as h3 after BN1
  float* hbn = ws;  ws += (size_t)N * D;
  float* tb  = ws;  ws += (size_t)N * 2 * D;
  float* h2  = ws;  ws += (size_t)N * D;
  float* st  = ws;  ws += 1024;
  float* Ee  = ws;                           // E * D

  float* sums1 = st;       float* sq1 = st + 128;
  float* sc1   = st + 256; float* sh1 = st + 384;
  float* sums2 = st + 512; float* sq2 = st + 640;
  float* sc2   = st + 768; float* sh2 = st + 896;

  hipMemsetAsync(wV, 0, (size_t)N * D  * sizeof(float), stream);
  hipMemsetAsync(Zb, 0, (size_t)N * NH * sizeof(float), stream);
  hipMemsetAsync(st, 0, 1024 * sizeof(float), stream);

  const int blk  = 32 * WAVES;               // 256 threads, 8 row-tiles per WG
  const int nBlk = (N / 16 + WAVES - 1) / WAVES;
  const int eBlk = (E / 16 + WAVES - 1) / WAVES;

  // projections
  gemm_wmma_f32<<<nBlk, blk, 0, stream>>>(x,  Wq, nullptr, Qb, N, D, D, 0, 0);
  gemm_wmma_f32<<<nBlk, blk, 0, stream>>>(x,  Wk, nullptr, Kb, N, D, D, 0, 0);
  gemm_wmma_f32<<<nBlk, blk, 0, stream>>>(x,  Wv, nullptr, Vb, N, D, D, 0, 0);
  gemm_wmma_f32<<<eBlk, blk, 0, stream>>>(ea, We, nullptr, Ee, E, D, D, 0, 0);

  // edge scoring + segment-sum scatter
  const long long tot = (long long)E * NH;
  edge_attn_scatter<<<(int)((tot + 255) / 256), 256, 0, stream>>>(
      Qb, Kb, Vb, Ee, ei, wV, Zb, E);

  // residual + BN1
  attn_residual_stats<<<1024, 128, 0, stream>>>(x, wV, Zb, h1, sums1, sq1, N);
  bn_finalize<<<1, 128, 0, stream>>>(sums1, sq1, g1, be1, sc1, sh1, N);
  bn_apply<<<2048, 256, 0, stream>>>(h1, sc1, sh1, hbn, (size_t)N * D);

  // FFN: relu(hbn @ W1 + b1) @ W2 + b2
  gemm_wmma_f32<<<nBlk, blk, 0, stream>>>(hbn, W1, b1, tb, N, D,     2 * D, 0,   1);
  gemm_wmma_f32<<<nBlk, blk, 0, stream>>>(hbn, W1, b1, tb, N, D,     2 * D, 128, 1);
  gemm_wmma_f32<<<nBlk, blk, 0, stream>>>(tb,  W2, b2, h2, N, 2 * D, D,     0,   0);

  // residual + BN2 -> output
  add_residual_stats<<<1024, 128, 0, stream>>>(hbn, h2, h1, sums2, sq2, N);
  bn_finalize<<<1, 128, 0, stream>>>(sums2, sq2, g2, be2, sc2, sh2, N);
  bn_apply<<<2048, 256, 0, stream>>>(h1, sc2, sh2, (float*)d_out, (size_t)N * D);
}